// SelectiveSSM_83897891160064
// MI455X (gfx1250) — compile-verified
//
#include <hip/hip_runtime.h>
#include <hip/hip_bf16.h>

// ---------------------------------------------------------------------------
// SelectiveSSM on MI455X (gfx1250, wave32).
//
// Pipeline:
//   K1 prep      : delta mean, dA = expm(A*mean) (scaling&squaring + Taylor),
//                  dA^128 by repeated squaring.             [1 block x 256]
//   K2 ub_gemm   : uB = u @ Bmat   (M=16384,K=2048,N=16)    WMMA f32 16x16x4
//   K3a scan     : per-chunk local scan, zero init -> chunk carries
//   K3b prefix   : sequential carry combine with dA^128 per batch
//   K3c scan     : local scan with correct init -> xs
//   K4 y_gemm    : y = (xs @ Cmat + u*D) / 64   (M=16384,K=16,N=2048)
//                                                           WMMA f32 16x16x4
// All f32 throughout (matches reference precision; products exact f32).
// ---------------------------------------------------------------------------

typedef __attribute__((ext_vector_type(2))) float v2f;
typedef __attribute__((ext_vector_type(8))) float v8f;

#define D_MODEL   1024
#define D_INNER   2048
#define D_STATE   16
#define DT_RANK   64
#define B_SZ      4
#define SEQ_LEN   4096
#define ROWS      (B_SZ * SEQ_LEN)     /* 16384 flat (b,l) rows */
#define CHUNK     128
#define NCHUNK_B  (SEQ_LEN / CHUNK)    /* 32 chunks per batch   */
#define NCHUNK    (B_SZ * NCHUNK_B)    /* 128 chunks total      */

// ---------------------------------------------------------------------------
// K1: delta mean -> expm(A*mean) -> dA, dA^128.  Single block, 256 threads.
// ---------------------------------------------------------------------------
__global__ __launch_bounds__(256)
void ssm_prep(const float* __restrict__ A,
              const float* __restrict__ dt_projs,
              const float* __restrict__ log_dt,
              float* __restrict__ dA_out,     // 256 floats
              float* __restrict__ dAC_out) {  // 256 floats (dA^128)
  __shared__ float edt[DT_RANK];
  __shared__ float red[256];
  __shared__ float M[256], T[256], E[256], term[256];
  __shared__ int   s_shift;
  const int tid = threadIdx.x;
  const int i = tid >> 4, j = tid & 15;

  if (tid < DT_RANK) edt[tid] = expf(log_dt[tid]);
  __syncthreads();

  // delta[d] = clip(exp(dt_projs[d,:] @ edt + 1e-6), 1e-6, 10); accumulate mean
  float lsum = 0.f;
  for (int d = tid; d < D_INNER; d += 256) {
    float s = 1e-6f;
    for (int r = 0; r < DT_RANK; ++r) s += dt_projs[d * DT_RANK + r] * edt[r];
    float delta = expf(s);
    delta = fminf(fmaxf(delta, 1e-6f), 10.0f);
    lsum += delta;
  }
  red[tid] = lsum;
  __syncthreads();
  for (int off = 128; off > 0; off >>= 1) {
    if (tid < off) red[tid] += red[tid + off];
    __syncthreads();
  }
  const float mean = red[0] * (1.0f / (float)D_INNER);
  __syncthreads();                       // everyone read red[0]

  M[tid] = A[tid] * mean;                // M = A * delta.mean()
  __syncthreads();

  // inf-norm for scaling
  if (tid < 16) {
    float rs = 0.f;
    for (int k = 0; k < 16; ++k) rs += fabsf(M[tid * 16 + k]);
    red[tid] = rs;
  }
  __syncthreads();
  if (tid == 0) {
    float nm = 0.f;
    for (int k = 0; k < 16; ++k) nm = fmaxf(nm, red[k]);
    int s = 0;
    while (nm > 0.25f && s < 40) { nm *= 0.5f; ++s; }
    s_shift = s;
  }
  __syncthreads();

  // T = M / 2^s ; Taylor: E = sum_{it} T^it / it!
  T[tid] = M[tid] * exp2f(-(float)s_shift);
  const float idv = (i == j) ? 1.f : 0.f;
  E[tid] = idv;
  term[tid] = idv;
  __syncthreads();
  for (int it = 1; it <= 16; ++it) {
    float sacc = 0.f;
    for (int k = 0; k < 16; ++k) sacc += term[i * 16 + k] * T[k * 16 + j];
    sacc *= (1.0f / (float)it);
    __syncthreads();
    term[tid] = sacc;
    E[tid] += sacc;
    __syncthreads();
  }
  // square back: E = E^(2^s)
  for (int sq = 0; sq < s_shift; ++sq) {
    float sacc = 0.f;
    for (int k = 0; k < 16; ++k) sacc += E[i * 16 + k] * E[k * 16 + j];
    __syncthreads();
    E[tid] = sacc;
    __syncthreads();
  }
  dA_out[tid] = E[tid];

  // dA^128 by 7 squarings
  T[tid] = E[tid];
  __syncthreads();
  for (int sq = 0; sq < 7; ++sq) {
    float sacc = 0.f;
    for (int k = 0; k < 16; ++k) sacc += T[i * 16 + k] * T[k * 16 + j];
    __syncthreads();
    T[tid] = sacc;
    __syncthreads();
  }
  dAC_out[tid] = T[tid];
}

// ---------------------------------------------------------------------------
// K2: uB[row, n] = sum_d u[row, d] * Bmat[d, n].
// One wave per 16-row M-tile; N = D_STATE = 16 exactly; K loop step 4 with
// V_WMMA_F32_16X16X4_F32.  A/B fragment layouts per CDNA5 ISA 7.12.2:
//   A (16x4): lane m = lane%16; lanes<16 hold K={k0,k0+1}, lanes>=16 K={k0+2,k0+3}
//   B (4x16): lane n = lane%16; same K split.
//   C/D     : vgpr r -> row r + (lane<16?0:8), col lane%16.
// ---------------------------------------------------------------------------
__global__ __launch_bounds__(256)
void ssm_ub_gemm(const float* __restrict__ u,
                 const float* __restrict__ Bm,
                 float* __restrict__ uB) {
  const int wave = (int)((blockIdx.x * 256u + threadIdx.x) >> 5);
  const int lane = (int)(threadIdx.x & 31u);
  const int m0   = wave * 16;
  const int nloc = lane & 15;
  const int koff = (lane < 16) ? 0 : 2;
  const float* __restrict__ urow = u + (size_t)(m0 + nloc) * D_INNER;

  v8f c = {};
  for (int k0 = 0; k0 < D_INNER; k0 += 4) {
    v2f a = *reinterpret_cast<const v2f*>(urow + k0 + koff);   // 8B aligned
    v2f b;
    b.x = Bm[(k0 + koff)     * D_STATE + nloc];
    b.y = Bm[(k0 + koff + 1) * D_STATE + nloc];
    c = __builtin_amdgcn_wmma_f32_16x16x4_f32(false, a, false, b,
                                              (short)0, c, false, false);
    if ((k0 & 31) == 0 && k0 + 64 < D_INNER)
      __builtin_prefetch(urow + k0 + 64, 0, 3);                // global_prefetch_b8
  }
#pragma unroll
  for (int r = 0; r < 8; ++r) {
    const int row = m0 + r + ((lane < 16) ? 0 : 8);
    uB[row * D_STATE + nloc] = c[r];
  }
}

// ---------------------------------------------------------------------------
// K3a/K3c: per-chunk local scan x_t = x_{t-1} @ dA + uB_t  (wave32, lanes 0..15
// own state elements; broadcast x[k] via __shfl).  init==null -> zero init.
// ---------------------------------------------------------------------------
__global__ __launch_bounds__(32)
void ssm_scan_chunks(const float* __restrict__ dA,
                     const float* __restrict__ uB,
                     const float* __restrict__ init,       // may be null
                     float* __restrict__ carry_out,        // may be null
                     float* __restrict__ xs_out) {         // may be null
  const int chunk = blockIdx.x;
  const int lane  = threadIdx.x;
  const bool act  = lane < 16;
  const int row0  = chunk * CHUNK;

  float col[16];
#pragma unroll
  for (int k = 0; k < 16; ++k) col[k] = act ? dA[k * 16 + lane] : 0.f;

  float x = (act && init) ? init[chunk * 16 + lane] : 0.f;

  for (int t = 0; t < CHUNK; ++t) {
    const float ub = act ? uB[(row0 + t) * D_STATE + lane] : 0.f;
    float p0 = ub, p1 = 0.f, p2 = 0.f, p3 = 0.f;
#pragma unroll
    for (int k = 0; k < 16; k += 4) {
      p0 = fmaf(__shfl(x, k,     32), col[k],     p0);
      p1 = fmaf(__shfl(x, k + 1, 32), col[k + 1], p1);
      p2 = fmaf(__shfl(x, k + 2, 32), col[k + 2], p2);
      p3 = fmaf(__shfl(x, k + 3, 32), col[k + 3], p3);
    }
    x = (p0 + p1) + (p2 + p3);
    if (xs_out && act) xs_out[(row0 + t) * D_STATE + lane] = x;
  }
  if (carry_out && act) carry_out[chunk * 16 + lane] = x;
}

// ---------------------------------------------------------------------------
// K3b: per batch, sequential prefix over chunk carries with dA^128:
//   start_0 = 0 ; start_{c+1} = start_c @ dA^128 + carry_c
// ---------------------------------------------------------------------------
__global__ __launch_bounds__(32)
void ssm_carry_prefix(const float* __restrict__ dAC,
                      const float* __restrict__ carries,
                      float* __restrict__ starts) {
  const int b    = blockIdx.x;
  const int lane = threadIdx.x;
  const bool act = lane < 16;

  float col[16];
#pragma unroll
  for (int k = 0; k < 16; ++k) col[k] = act ? dAC[k * 16 + lane] : 0.f;

  float x = 0.f;
  for (int c = 0; c < NCHUNK_B; ++c) {
    const int idx = (b * NCHUNK_B + c) * 16 + lane;
    if (act) starts[idx] = x;
    const float f = act ? carries[idx] : 0.f;
    float p0 = f, p1 = 0.f, p2 = 0.f, p3 = 0.f;
#pragma unroll
    for (int k = 0; k < 16; k += 4) {
      p0 = fmaf(__shfl(x, k,     32), col[k],     p0);
      p1 = fmaf(__shfl(x, k + 1, 32), col[k + 1], p1);
      p2 = fmaf(__shfl(x, k + 2, 32), col[k + 2], p2);
      p3 = fmaf(__shfl(x, k + 3, 32), col[k + 3], p3);
    }
    x = (p0 + p1) + (p2 + p3);
  }
}

// ---------------------------------------------------------------------------
// K4: y[row, d] = (sum_n xs[row, n] * Cmat[n, d] + u[row, d]*D[d]) / 64.
// One wave per 16-row M-tile, looping 128 N-tiles of 16; K=16 -> 4 WMMAs.
// A fragments (xs tile) loaded once, reused for all N-tiles.
// ---------------------------------------------------------------------------
__global__ __launch_bounds__(256)
void ssm_y_gemm(const float* __restrict__ xs,
                const float* __restrict__ Cm,
                const float* __restrict__ u,
                const float* __restrict__ Dv,
                float* __restrict__ y) {
  const int wave = (int)((blockIdx.x * 256u + threadIdx.x) >> 5);
  const int lane = (int)(threadIdx.x & 31u);
  const int m0   = wave * 16;
  const int l16  = lane & 15;
  const int koff = (lane < 16) ? 0 : 2;
  const float inv = 0.015625f;  // 1/sqrt(4096)

  v2f a[4];
  const float* __restrict__ xr = xs + (size_t)(m0 + l16) * D_STATE;
#pragma unroll
  for (int kk = 0; kk < 4; ++kk)
    a[kk] = *reinterpret_cast<const v2f*>(xr + kk * 4 + koff);

  for (int n0 = 0; n0 < D_INNER; n0 += 16) {
    v8f c = {};
#pragma unroll
    for (int kk = 0; kk < 4; ++kk) {
      v2f b;
      b.x = Cm[(kk * 4 + koff)     * D_INNER + n0 + l16];
      b.y = Cm[(kk * 4 + koff + 1) * D_INNER + n0 + l16];
      c = __builtin_amdgcn_wmma_f32_16x16x4_f32(false, a[kk], false, b,
                                                (short)0, c, false, false);
    }
    const float dvv = Dv[n0 + l16];
#pragma unroll
    for (int r = 0; r < 8; ++r) {
      const int row  = m0 + r + ((lane < 16) ? 0 : 8);
      const size_t idx = (size_t)row * D_INNER + n0 + l16;
      y[idx] = (c[r] + u[idx] * dvv) * inv;
    }
  }
}

// ---------------------------------------------------------------------------
// Launch.  Workspace layout (floats):
//   [0,256)       dA
//   [256,512)     dA^128
//   [512,2560)    chunk carries   (128*16)
//   [2560,4608)   chunk starts    (128*16)
//   [4608,266752) uB              (16384*16)
//   [266752,528896) xs            (16384*16)
// Total ~2.1 MB.
// ---------------------------------------------------------------------------
extern "C" void kernel_launch(void* const* d_in, const int* in_sizes, int n_in,
                              void* d_out, int out_size, void* d_ws, size_t ws_size,
                              hipStream_t stream) {
  (void)in_sizes; (void)n_in; (void)out_size; (void)ws_size;
  const float* u   = (const float*)d_in[0];
  const float* A   = (const float*)d_in[1];
  const float* Bm  = (const float*)d_in[2];
  const float* Cm  = (const float*)d_in[3];
  const float* Dv  = (const float*)d_in[4];
  const float* dtp = (const float*)d_in[5];
  const float* ldt = (const float*)d_in[6];
  float* out = (float*)d_out;
  float* ws  = (float*)d_ws;

  float* dA      = ws;
  float* dAC     = ws + 256;
  float* carries = ws + 512;
  float* starts  = ws + 512 + NCHUNK * 16;
  float* uB      = ws + 512 + 2 * NCHUNK * 16;
  float* xs      = uB + (size_t)ROWS * D_STATE;

  ssm_prep<<<1, 256, 0, stream>>>(A, dtp, ldt, dA, dAC);
  ssm_ub_gemm<<<ROWS / 16 / 8, 256, 0, stream>>>(u, Bm, uB);
  ssm_scan_chunks<<<NCHUNK, 32, 0, stream>>>(dA, uB, nullptr, carries, nullptr);
  ssm_carry_prefix<<<B_SZ, 32, 0, stream>>>(dAC, carries, starts);
  ssm_scan_chunks<<<NCHUNK, 32, 0, stream>>>(dA, uB, starts, nullptr, xs);
  ssm_y_gemm<<<ROWS / 16 / 8, 256, 0, stream>>>(xs, Cm, u, Dv, out);
}